// GCN_78589311582297
// MI455X (gfx1250) — compile-verified
//
#include <hip/hip_runtime.h>
#include <hip/hip_bf16.h>

typedef __attribute__((ext_vector_type(2))) float v2f;
typedef __attribute__((ext_vector_type(8))) float v8f;

// ---------------------------------------------------------------------------
// Degree / normalization kernels
// ---------------------------------------------------------------------------
__global__ void k_init_deg(float* __restrict__ deg, int n) {
  int i = blockIdx.x * blockDim.x + threadIdx.x;
  if (i < n) deg[i] = 1.0f;  // self-loop contributes 1 to every node's degree
}

__global__ void k_count_deg(const int* __restrict__ dst, float* __restrict__ deg, int e) {
  int i = blockIdx.x * blockDim.x + threadIdx.x;
  if (i < e) atomicAdd(&deg[dst[i]], 1.0f);  // exact: counts < 2^24
}

__global__ void k_rsqrt_inplace(float* __restrict__ d, int n) {
  int i = blockIdx.x * blockDim.x + threadIdx.x;
  if (i < n) d[i] = rsqrtf(d[i]);  // deg >= 1 always (self-loops)
}

// ---------------------------------------------------------------------------
// Dense GEMM: C[M,N] = A[M,K] @ B[K,N], fp32-exact via V_WMMA_F32_16X16X4_F32.
// One wave (32 threads) computes one 16x16 tile of C; K fully unrolled
// (K/4 straight-line WMMAs). M must be a multiple of 16 (100000 % 16 == 0).
//
// N=40 tail handling: the B column index is CLAMPED (never predicated), so
// control flow stays uniform through the WMMA loop (EXEC all-ones, no
// saveexec dance); out-of-range lanes compute duplicate columns that are
// simply never stored.
//
// ISA lane layouts (CDNA5 ISA):
//   A 16x4 f32 : lanes 0-15 -> M=lane, VGPR0=K0, VGPR1=K1; lanes 16-31 -> K2,K3
//   B 4x16 f32 : lanes 0-15 -> N=lane, K rows 0/1; lanes 16-31 -> K rows 2/3
//   C 16x16 f32: lane 0-15 -> N=lane, VGPR r -> M=r; lanes 16-31 -> M=r+8
// ---------------------------------------------------------------------------
template <int K, int N>
__global__ void k_gemm_wmma_f32(const float* __restrict__ A,
                                const float* __restrict__ B,
                                float* __restrict__ C) {
  const int lane = threadIdx.x & 31;
  const int row  = lane & 15;   // A row within tile / B-C column within tile
  const int kp   = lane >> 4;   // K-pair select (0 or 1)
  const int m0   = blockIdx.x * 16;
  const int n0   = blockIdx.y * 16;
  const int col  = n0 + row;
  const int colc = (col < N) ? col : (N - 1);  // clamped, always-in-bounds

  const float* Arow = A + (size_t)(m0 + row) * K;
  v8f c = {};
#pragma unroll
  for (int k0 = 0; k0 < K; k0 += 4) {
    const int ka = k0 + 2 * kp;
    v2f a, b;
    a.x = Arow[ka];
    a.y = Arow[ka + 1];
    b.x = B[(size_t)ka * N + colc];
    b.y = B[(size_t)(ka + 1) * N + colc];
    // 8 args: (neg_a, A, neg_b, B, c_mod, C, reuse_a, reuse_b)
    c = __builtin_amdgcn_wmma_f32_16x16x4_f32(false, a, false, b, (short)0, c,
                                              false, false);
  }
  if (col < N) {
#pragma unroll
    for (int r = 0; r < 8; ++r)
      C[(size_t)(m0 + r + 8 * kp) * N + col] = c[r];
  }
}

// ---------------------------------------------------------------------------
// acc[i,f] = h[i,f] * dinv[i]^2   (the self-loop edge: src==dst==i)
// ---------------------------------------------------------------------------
__global__ void k_self_loop_init(const float* __restrict__ h,
                                 const float* __restrict__ dinv,
                                 float* __restrict__ acc, int n, int stride) {
  int idx = blockIdx.x * blockDim.x + threadIdx.x;
  if (idx >= n * stride) return;
  int i = idx / stride;
  float di = dinv[i];
  acc[idx] = h[idx] * di * di;
}

// ---------------------------------------------------------------------------
// Edge scatter-add: acc[dst] += h[src] * dinv[src]*dinv[dst]
// F4 float4-chunks per edge; one thread per (edge, chunk). h and acc are
// L2-resident (25.6MB / 16MB << 192MB L2), so this is L2-atomic bound.
// ---------------------------------------------------------------------------
template <int F4, int STRIDE>
__global__ void k_edge_agg(const int* __restrict__ src, const int* __restrict__ dst,
                           const float* __restrict__ dinv,
                           const float* __restrict__ h,
                           float* __restrict__ acc, int e) {
  int tid = blockIdx.x * blockDim.x + threadIdx.x;
  if (tid >= e * F4) return;
  int ei  = tid / F4;
  int sub = tid - ei * F4;
  int s = src[ei], d = dst[ei];
  float norm = dinv[s] * dinv[d];
  const float4 v = *(const float4*)(h + (size_t)s * STRIDE + sub * 4);
  float* ap = acc + (size_t)d * STRIDE + sub * 4;
  atomicAdd(ap + 0, v.x * norm);
  atomicAdd(ap + 1, v.y * norm);
  atomicAdd(ap + 2, v.z * norm);
  atomicAdd(ap + 3, v.w * norm);
}

// ---------------------------------------------------------------------------
// h1 = relu(acc + b)
// ---------------------------------------------------------------------------
__global__ void k_relu_bias(const float* __restrict__ acc,
                            const float* __restrict__ b,
                            float* __restrict__ out, int n, int stride) {
  int idx = blockIdx.x * blockDim.x + threadIdx.x;
  if (idx >= n * stride) return;
  int f = idx % stride;
  out[idx] = fmaxf(acc[idx] + b[f], 0.0f);
}

// ---------------------------------------------------------------------------
// out[i,:] = log_softmax(out[i,:] + b2), 40 classes, one thread per node.
// ---------------------------------------------------------------------------
__global__ void k_bias_logsoftmax40(float* __restrict__ out,
                                    const float* __restrict__ b, int n) {
  int i = blockIdx.x * blockDim.x + threadIdx.x;
  if (i >= n) return;
  float* p = out + (size_t)i * 40;
  float v[40];
  float m = -3.402823466e38f;
#pragma unroll
  for (int f = 0; f < 40; ++f) {
    v[f] = p[f] + b[f];
    m = fmaxf(m, v[f]);
  }
  float s = 0.0f;
#pragma unroll
  for (int f = 0; f < 40; ++f) s += expf(v[f] - m);
  float lse = m + logf(s);
#pragma unroll
  for (int f = 0; f < 40; ++f) p[f] = v[f] - lse;
}

// ---------------------------------------------------------------------------
// Launch
// ---------------------------------------------------------------------------
extern "C" void kernel_launch(void* const* d_in, const int* in_sizes, int n_in,
                              void* d_out, int out_size, void* d_ws, size_t ws_size,
                              hipStream_t stream) {
  (void)n_in; (void)out_size; (void)ws_size;

  const float* x  = (const float*)d_in[0];   // [N,128]
  const int*   ei = (const int*)d_in[1];     // [2,E]
  const float* W1 = (const float*)d_in[2];   // [128,64]
  const float* b1 = (const float*)d_in[3];   // [64]
  const float* W2 = (const float*)d_in[4];   // [64,40]
  const float* b2 = (const float*)d_in[5];   // [40]
  float* out = (float*)d_out;                // [N,40]

  const int F_IN = 128, F_H = 64, F_OUT = 40;
  const int Nn = in_sizes[0] / F_IN;   // 100000 (multiple of 16)
  const int E  = in_sizes[1] / 2;      // 800000
  const int* src = ei;
  const int* dst = ei + E;

  // Workspace layout (all fp32):
  //   [0,   4MB)  dinv : Nn floats
  //   [4,  36MB)  h    : Nn*64 floats (layer-1 GEMM out, then reused for h1)
  //   [36, 68MB)  acc1 : Nn*64 floats
  //   [68, 88MB)  h2   : Nn*40 floats
  char* ws = (char*)d_ws;
  float* dinv = (float*)(ws);
  float* h    = (float*)(ws + ((size_t)4  << 20));
  float* acc1 = (float*)(ws + ((size_t)36 << 20));
  float* h2   = (float*)(ws + ((size_t)68 << 20));

  const dim3 B256(256);

  // --- symmetric normalization coefficients -------------------------------
  k_init_deg<<<(Nn + 255) / 256, B256, 0, stream>>>(dinv, Nn);
  k_count_deg<<<(E + 255) / 256, B256, 0, stream>>>(dst, dinv, E);
  k_rsqrt_inplace<<<(Nn + 255) / 256, B256, 0, stream>>>(dinv, Nn);

  // --- layer 1: h = x @ W1 ; aggregate ; relu(+b1) ------------------------
  k_gemm_wmma_f32<128, 64><<<dim3(Nn / 16, 4), dim3(32), 0, stream>>>(x, W1, h);
  k_self_loop_init<<<((Nn * F_H) + 255) / 256, B256, 0, stream>>>(h, dinv, acc1, Nn, F_H);
  k_edge_agg<16, 64><<<((E * 16) + 255) / 256, B256, 0, stream>>>(src, dst, dinv, h, acc1, E);
  k_relu_bias<<<((Nn * F_H) + 255) / 256, B256, 0, stream>>>(acc1, b1, h, Nn, F_H);

  // --- layer 2: h2 = h1 @ W2 ; aggregate into d_out ; +b2 ; log_softmax ---
  k_gemm_wmma_f32<64, 40><<<dim3(Nn / 16, 3), dim3(32), 0, stream>>>(h, W2, h2);
  k_self_loop_init<<<((Nn * F_OUT) + 255) / 256, B256, 0, stream>>>(h2, dinv, out, Nn, F_OUT);
  k_edge_agg<10, 40><<<((E * 10) + 255) / 256, B256, 0, stream>>>(src, dst, dinv, h2, out, E);
  k_bias_logsoftmax40<<<(Nn + 255) / 256, B256, 0, stream>>>(out, b2, Nn);
}